// EffiGlobalExtractor_13254269075923
// MI455X (gfx1250) — compile-verified
//
#include <hip/hip_runtime.h>
#include <math.h>

#define NEGV 1000000000000.0f

typedef __attribute__((ext_vector_type(2))) float v2f;
typedef __attribute__((ext_vector_type(8))) float v8f;

// -------------------------------------------------------------------------
// O[M,N] = X[M,K] @ W[K,N] + bias[N]  via V_WMMA_F32_16X16X4_F32 (wave32)
// One wave (32 threads) per 16x16 output tile.
// A 16x4 layout: lane -> M = lane&15, VGPR v -> K = v + 2*(lane>>4)
// B 4x16 layout: lane -> N = lane&15, VGPR v -> K = v + 2*(lane>>4)
// C/D 16x16:     VGPR r -> M = r + 8*(lane>>4), N = lane&15
// -------------------------------------------------------------------------
__global__ __launch_bounds__(32) void gemm_xw_bias(const float* __restrict__ X,
                                                   const float* __restrict__ W,
                                                   const float* __restrict__ bias,
                                                   float* __restrict__ O,
                                                   int M, int N, int K) {
  const int lane = threadIdx.x;
  const int idx16 = lane & 15;
  const int half  = lane >> 4;
  const int ntn = N >> 4;
  const int mt = blockIdx.x / ntn, nt = blockIdx.x % ntn;
  const int m0 = mt << 4, n0 = nt << 4;

  const float* Xp = X + (size_t)(m0 + idx16) * K + 2 * half;
  const float* Wp = W + (size_t)(2 * half) * N + n0 + idx16;

  v8f acc = {};
  for (int k0 = 0; k0 < K; k0 += 4) {
    v2f a, b;
    a.x = Xp[k0];
    a.y = Xp[k0 + 1];
    b.x = Wp[(size_t)k0 * N];
    b.y = Wp[(size_t)(k0 + 1) * N];
    acc = __builtin_amdgcn_wmma_f32_16x16x4_f32(false, a, false, b,
                                                (short)0, acc, false, false);
  }
  const float bn = bias[n0 + idx16];
#pragma unroll
  for (int r = 0; r < 8; ++r) {
    const int m = m0 + r + 8 * half;
    O[(size_t)m * N + n0 + idx16] = acc[r] + bn;
  }
}

// -------------------------------------------------------------------------
// Split P(B*S,128) into rope(qw), rope(kw); fold the 1/sqrt(64) into Q.
// qw[d] = P[2d], kw[d] = P[2d+1]; rope pairs (2j, 2j+1).
// One 32-thread block per (b,s) row; thread j handles rotation pair j.
// -------------------------------------------------------------------------
__global__ __launch_bounds__(32) void rope_split(const float* __restrict__ P,
                                                 float* __restrict__ Q,
                                                 float* __restrict__ Kr,
                                                 int S) {
  const int bs = blockIdx.x;
  const int s = bs % S;
  const int j = threadIdx.x;  // 0..31
  const float4 p = ((const float4*)(P + (size_t)bs * 128))[j];
  // inv = 10000^(-j/32)
  const float inv = expf(-(float)j * (9.210340371976184f / 32.0f));
  float sn, c;
  sincosf((float)s * inv, &sn, &c);
  const float q0 = p.x, k0 = p.y, q1 = p.z, k1 = p.w;
  float* Qp = Q + (size_t)bs * 64 + 2 * j;
  float* Kp = Kr + (size_t)bs * 64 + 2 * j;
  Qp[0] = (q0 * c - q1 * sn) * 0.125f;
  Qp[1] = (q1 * c + q0 * sn) * 0.125f;
  Kp[0] = k0 * c - k1 * sn;
  Kp[1] = k1 * c + k0 * sn;
}

// -------------------------------------------------------------------------
// biasT[b, c, s] = (hidden[b,s,:] . W2[:,c] + b2[c]) * 0.5   (C = 2T small)
// One block per (b,s); hidden row staged in LDS.
// -------------------------------------------------------------------------
__global__ __launch_bounds__(32) void bias_head(const float* __restrict__ X,
                                                const float* __restrict__ W2,
                                                const float* __restrict__ b2,
                                                float* __restrict__ biasT,
                                                int S, int K, int C) {
  __shared__ float xrow[768];
  const int bs = blockIdx.x;
  for (int k = threadIdx.x; k < K; k += 32) xrow[k] = X[(size_t)bs * K + k];
  __syncthreads();
  const int c = threadIdx.x;
  if (c < C) {
    float acc = b2[c];
    for (int k = 0; k < K; ++k) acc += xrow[k] * W2[(size_t)k * C + c];
    const int b = bs / S, s = bs % S;
    biasT[((size_t)b * C + c) * S + s] = acc * 0.5f;
  }
}

// -------------------------------------------------------------------------
// Fused: qk 16x16 tile via WMMA (K=64), then for each type channel t add
// biases, apply mask + strict-lower-triangle penalty, and stream the stores.
// out[b,t,m,n] with qk already scaled by 0.125 (folded into Q).
// -------------------------------------------------------------------------
__global__ __launch_bounds__(32) void pointer_logits(const float* __restrict__ Q,
                                                     const float* __restrict__ Kr,
                                                     const float* __restrict__ biasT,
                                                     const int* __restrict__ maskI,
                                                     float* __restrict__ out,
                                                     int S, int T) {
  const int lane = threadIdx.x;
  const int idx16 = lane & 15;
  const int half  = lane >> 4;
  const int nt16 = S >> 4;  // 32
  int tile = blockIdx.x;
  const int b = tile / (nt16 * nt16);
  tile -= b * nt16 * nt16;
  const int m0 = (tile / nt16) << 4;
  const int n0 = (tile % nt16) << 4;
  const int ncol = n0 + idx16;

  const float* Qp = Q + ((size_t)(b * S + m0 + idx16)) * 64 + 2 * half;
  const float* Kp = Kr + ((size_t)(b * S + ncol)) * 64 + 2 * half;

  v8f acc = {};
#pragma unroll
  for (int k0 = 0; k0 < 64; k0 += 4) {
    v2f a, bb;
    a.x = Qp[k0];
    a.y = Qp[k0 + 1];
    bb.x = Kp[k0];
    bb.y = Kp[k0 + 1];
    acc = __builtin_amdgcn_wmma_f32_16x16x4_f32(false, a, false, bb,
                                                (short)0, acc, false, false);
  }

  const float mk = (float)maskI[b * S + ncol];
  float mq[8], tr[8];
  int mr[8];
#pragma unroll
  for (int r = 0; r < 8; ++r) {
    mr[r] = m0 + r + 8 * half;
    mq[r] = (float)maskI[b * S + mr[r]];
    tr[r] = (mr[r] > ncol) ? NEGV : 0.0f;
  }

  const int C = 2 * T;
  for (int t = 0; t < T; ++t) {
    const float bk = biasT[((size_t)b * C + 2 * t) * S + ncol];
    const float* bqp = biasT + ((size_t)b * C + 2 * t + 1) * S;
    float* op = out + ((size_t)(b * T + t) * S) * S + ncol;
#pragma unroll
    for (int r = 0; r < 8; ++r) {
      float v = acc[r] + bk + bqp[mr[r]];
      v = v * mq[r] - NEGV * (1.0f - mq[r]);
      v = v * mk - NEGV * (1.0f - mk);
      v -= tr[r];
      op[(size_t)mr[r] * S] = v;
    }
  }
}

extern "C" void kernel_launch(void* const* d_in, const int* in_sizes, int n_in,
                              void* d_out, int out_size, void* d_ws, size_t ws_size,
                              hipStream_t stream) {
  const float* hidden = (const float*)d_in[0];  // (8,512,768) f32
  const int*   mask   = (const int*)d_in[1];    // (8,512) i32
  const float* W1  = (const float*)d_in[2];
  const float* b1  = (const float*)d_in[3];
  const float* W2  = (const float*)d_in[4];
  const float* b2  = (const float*)d_in[5];
  const float* Wa1 = (const float*)d_in[6];
  const float* ba1 = (const float*)d_in[7];
  const float* Wa2 = (const float*)d_in[8];
  const float* ba2 = (const float*)d_in[9];

  const int B = 8, S = 512, H = 768, D2 = 128;
  const int entT = in_sizes[5] / 2;  // 12
  const int argT = in_sizes[9] / 2;  // 16
  const int M = B * S;               // 4096

  float* ws  = (float*)d_ws;
  float* P   = ws;                              // M*128 (reused by both heads)
  float* Q1  = P  + (size_t)M * 128;            // M*64
  float* K1  = Q1 + (size_t)M * 64;
  float* Q2  = K1 + (size_t)M * 64;
  float* K2  = Q2 + (size_t)M * 64;
  float* bT1 = K2 + (size_t)M * 64;             // B*2*entT*S
  float* bT2 = bT1 + (size_t)B * 2 * entT * S;  // B*2*argT*S

  float* out1 = (float*)d_out;
  float* out2 = out1 + (size_t)B * entT * S * S;

  const dim3 w32(32);
  const int gemmTiles = (M / 16) * (D2 / 16);  // 2048 waves
  const int ptrTiles  = B * (S / 16) * (S / 16);  // 8192 waves

  // head 1 projection + rope
  gemm_xw_bias<<<gemmTiles, w32, 0, stream>>>(hidden, W1, b1, P, M, D2, H);
  rope_split<<<M, w32, 0, stream>>>(P, Q1, K1, S);
  // head 2 projection + rope (reuses P)
  gemm_xw_bias<<<gemmTiles, w32, 0, stream>>>(hidden, Wa1, ba1, P, M, D2, H);
  rope_split<<<M, w32, 0, stream>>>(P, Q2, K2, S);
  // small bias GEMMs (transposed output, /2 folded in)
  bias_head<<<M, w32, 0, stream>>>(hidden, W2, b2, bT1, S, H, 2 * entT);
  bias_head<<<M, w32, 0, stream>>>(hidden, Wa2, ba2, bT2, S, H, 2 * argT);
  // fused qk-WMMA + broadcast/mask/tril + output streaming
  pointer_logits<<<ptrTiles, w32, 0, stream>>>(Q1, K1, bT1, mask, out1, S, entT);
  pointer_logits<<<ptrTiles, w32, 0, stream>>>(Q2, K2, bT2, mask, out2, S, argT);
}